// LabelAwarePool_63247688401687
// MI455X (gfx1250) — compile-verified
//
#include <hip/hip_runtime.h>
#include <hip/hip_bf16.h>
#include <stdint.h>

// Problem constants from the reference: D edges per destination node, keep D/2.
#define D_DEG 64
#define K_KEEP (D_DEG / 2)          // 32
#define WPB 8                       // one wave32 per node, 8 nodes per block
#define BLOCK_THREADS (WPB * 32)    // 256

// Low 32 bits of a generic pointer to a __shared__ object are the
// workgroup-relative LDS byte offset (flat LDS aperture keeps the offset in
// bits [31:0]), which is exactly what the async-to/from-LDS instructions take.
__device__ __forceinline__ unsigned lds_off_u32(const void* p) {
  return (unsigned)(uintptr_t)p;
}

__global__ __launch_bounds__(BLOCK_THREADS)
void label_pool_topk_kernel(const int* __restrict__ eSrc,
                            const int* __restrict__ eDst,
                            const float* __restrict__ sig,
                            int* __restrict__ outSrc,
                            int* __restrict__ outDst) {
  __shared__ alignas(16) int sSrc[WPB * D_DEG];                   // 2 KB
  __shared__ alignas(16) int sDst[WPB * D_DEG];                   // 2 KB
  __shared__ alignas(16) unsigned long long sKey[WPB * D_DEG];    // 4 KB
  __shared__ alignas(16) int sSelSrc[WPB * K_KEEP];               // 1 KB
  __shared__ alignas(16) int sSelDst[WPB * K_KEEP];               // 1 KB

  const int t    = threadIdx.x;
  const int w    = t >> 5;    // wave id == node slot in block
  const int lane = t & 31;

  // ---- Stage 1: async DMA this block's 512 src + 512 dst indices into LDS.
  // 128 threads x 16B per array; tracked by ASYNCcnt (CDNA5 async-to-LDS path).
  {
    const unsigned sub  = (unsigned)(t & 127);
    const unsigned goff = blockIdx.x * (WPB * D_DEG * 4u) + sub * 16u; // bytes
    if (t < 128) {
      unsigned la = lds_off_u32(&sSrc[sub * 4]);
      asm volatile("global_load_async_to_lds_b128 %0, %1, %2"
                   :: "v"(la), "v"(goff), "s"(eSrc) : "memory");
    } else {
      unsigned la = lds_off_u32(&sDst[sub * 4]);
      asm volatile("global_load_async_to_lds_b128 %0, %1, %2"
                   :: "v"(la), "v"(goff), "s"(eDst) : "memory");
    }
  }
  asm volatile("s_wait_asynccnt 0" ::: "memory");
  __syncthreads();

  // ---- Stage 2: per-lane similarities and exact order keys.
  // Lane owns segment elements i0 = 2*lane and i1 = 2*lane+1 (contiguous, so
  // edge reads are single ds_load_b64 and the key store is one ds_store_b128).
  const int nb = w * D_DEG;
  const int2 sp = ((const int2*)&sSrc[nb])[lane];   // {src[2l], src[2l+1]}
  const int2 dp = ((const int2*)&sDst[nb])[lane];   // {dst[2l], dst[2l+1]}

  const float a0 = sig[sp.x];
  const float a1 = sig[sp.y];
  const float b0 = sig[dp.x];
  const float b1 = sig[dp.y];

  const float sim0 = 1.0f - fabsf(a0 - b0);
  const float sim1 = 1.0f - fabsf(a1 - b1);

  // sim >= 0 -> IEEE bits are order-monotone. Append (63 - idx) so equal sims
  // rank the lower in-segment index first (jax.lax.top_k tie-break). Keys are
  // unique, so rank = #{keys > mine} is this element's exact sorted position.
  const unsigned long long k0 =
      ((unsigned long long)__float_as_uint(sim0) << 6) |
      (unsigned long long)(63 - 2 * lane);
  const unsigned long long k1 =
      ((unsigned long long)__float_as_uint(sim1) << 6) |
      (unsigned long long)(62 - 2 * lane);          // 63 - (2*lane + 1)

  ((ulonglong2*)&sKey[nb])[lane] = make_ulonglong2(k0, k1);
  __syncthreads();

  // ---- Stage 3: all-pairs rank, one v_cmp_gt_u64 per compare, 16B LDS
  // broadcasts (all lanes read the same address -> conflict-free).
  unsigned r0 = 0, r1 = 0;
  const ulonglong2* kp = (const ulonglong2*)&sKey[nb];
#pragma unroll
  for (int j = 0; j < D_DEG / 2; ++j) {
    const ulonglong2 kk = kp[j];
    r0 += (unsigned)(kk.x > k0) + (unsigned)(kk.y > k0);
    r1 += (unsigned)(kk.x > k1) + (unsigned)(kk.y > k1);
  }

  // Ranks are a permutation of 0..63: exactly K_KEEP winners per node, and the
  // rank is the descending-sorted output slot.
  if (r0 < (unsigned)K_KEEP) {
    sSelSrc[w * K_KEEP + r0] = sp.x;
    sSelDst[w * K_KEEP + r0] = dp.x;
  }
  if (r1 < (unsigned)K_KEEP) {
    sSelSrc[w * K_KEEP + r1] = sp.y;
    sSelDst[w * K_KEEP + r1] = dp.y;
  }
  __syncthreads();

  // ---- Stage 4: coalesced async DMA of the compacted winners LDS -> global.
  // 64 threads x 16B per output row (8 nodes x 32 kept x 4B = 1 KB per row).
  if (t < 64) {
    unsigned la   = lds_off_u32(&sSelSrc[t * 4]);
    unsigned goff = blockIdx.x * (WPB * K_KEEP * 4u) + (unsigned)t * 16u;
    asm volatile("global_store_async_from_lds_b128 %0, %1, %2"
                 :: "v"(goff), "v"(la), "s"(outSrc) : "memory");
  } else if (t < 128) {
    unsigned q    = (unsigned)(t - 64);
    unsigned la   = lds_off_u32(&sSelDst[q * 4]);
    unsigned goff = blockIdx.x * (WPB * K_KEEP * 4u) + q * 16u;
    asm volatile("global_store_async_from_lds_b128 %0, %1, %2"
                 :: "v"(goff), "v"(la), "s"(outDst) : "memory");
  }
  // s_endpgm performs an implicit wait-idle (drains ASYNCcnt) before retiring.
}

__global__ void sigmoid_kernel(const float* __restrict__ x,
                               float* __restrict__ s, int n) {
  int i = blockIdx.x * blockDim.x + threadIdx.x;
  if (i < n) {
    float v = x[i];
    s[i] = 1.0f / (1.0f + expf(-v));
  }
}

extern "C" void kernel_launch(void* const* d_in, const int* in_sizes, int n_in,
                              void* d_out, int out_size, void* d_ws, size_t ws_size,
                              hipStream_t stream) {
  const float* logits = (const float*)d_in[0];
  const int*   edge   = (const int*)d_in[1];

  const int N = in_sizes[0];            // nodes
  const int E = in_sizes[1] / 2;        // edges (edge_index is [2, E])
  const int* eSrc = edge;               // row 0
  const int* eDst = edge + E;           // row 1

  float* sig = (float*)d_ws;            // N floats of sigmoid(logits)

  sigmoid_kernel<<<(N + 255) / 256, 256, 0, stream>>>(logits, sig, N);

  int* outSrc = (int*)d_out;
  int* outDst = outSrc + (size_t)(out_size / 2);   // row 1 of [2, N*K]

  const int nNodes = E / D_DEG;         // == N for this problem
  const int blocks = nNodes / WPB;      // 262144 / 8 = 32768
  label_pool_topk_kernel<<<blocks, BLOCK_THREADS, 0, stream>>>(
      eSrc, eDst, sig, outSrc, outDst);
}